// IIR2DBank_11089605558743
// MI455X (gfx1250) — compile-verified
//
#include <hip/hip_runtime.h>
#include <stdint.h>

// CDNA5 / gfx1250. Wave32. WMMA-based blocked IIR scan + async global<->LDS DMA.
typedef __attribute__((ext_vector_type(2))) float v2f;
typedef __attribute__((ext_vector_type(8))) float v8f;

#define NN   4
#define HH   256
#define WW   256
#define CCH  96
#define NCC  6        // 6 chunks of 16 channels
#define TILE 4096     // 256 positions * 16 channels (floats)

// ---------------------------------------------------------------------------
// Async global<->LDS DMA (ASYNCcnt-tracked), GVS addressing:
//   mem_addr = SGPR_base + VGPR_i32_offset. LDS address = low 32 bits of the
//   generic shared-memory pointer (aperture lives in the high 32 bits).
// ---------------------------------------------------------------------------
__device__ __forceinline__ uint32_t lds_addr(const void* p) {
    return (uint32_t)(uintptr_t)p;
}
__device__ __forceinline__ void async_load_b128(uint32_t lds, const void* sbase,
                                                uint32_t voff) {
    asm volatile("global_load_async_to_lds_b128 %0, %1, %2"
                 :: "v"(lds), "v"(voff), "s"(sbase) : "memory");
}
__device__ __forceinline__ void async_store_b128(uint32_t lds, void* sbase,
                                                 uint32_t voff) {
    asm volatile("global_store_async_from_lds_b128 %0, %1, %2"
                 :: "v"(voff), "v"(lds), "s"(sbase) : "memory");
}
__device__ __forceinline__ void wait_async0() {
    asm volatile("s_wait_asynccnt 0" ::: "memory");
}

// a^e for integer e in [0,16] via square-and-multiply selects (~8 VALU, no powf).
__device__ __forceinline__ float pow_int(float a, float a2, float a4, float a8,
                                         float a16, int e) {
    float r = (e & 1) ? a : 1.0f;
    r *= (e & 2) ? a2 : 1.0f;
    r *= (e & 4) ? a4 : 1.0f;
    r *= (e & 8) ? a8 : 1.0f;
    r = (e & 16) ? r * a16 : r;
    return r;
}

// Per-filter WMMA operands for the blocked scan.
// Forward:  y[16] = T * x[16] + p*carry,  T[i][j] = (1-a) a^(i-j) (i>=j), p_i = a^(i+1)
// Backward: r[16] = U * z[16] + q*carry,  U[i][j] = (1-a) a^(j-i) (j>=i), q_i = a^(16-i)
struct FiltOps {
    v2f   AF[4];   // forward  T, split into 4 K=4 column chunks (A-matrix layout)
    v2f   AB[4];   // backward U
    float PF[8];   // forward carry coeff per D-VGPR row (m = r + 8*half)
    float QB[8];   // backward carry coeff
};

__device__ __forceinline__ FiltOps make_ops(float a, int lane) {
    FiltOps o;
    const float onem = 1.0f - a;
    const float a2 = a * a, a4 = a2 * a2, a8 = a4 * a4, a16 = a8 * a8;
    const int half = lane >> 4;
    const int m    = lane & 15;       // A-matrix row for this lane
#pragma unroll
    for (int k = 0; k < 4; ++k) {
        // A 16x4 f32 layout: lanes 0-15 hold K=0,1 ; lanes 16-31 hold K=2,3
        const int j0 = 4 * k + 2 * half;
        const int j1 = j0 + 1;
        o.AF[k].x = (m >= j0) ? onem * pow_int(a, a2, a4, a8, a16, m - j0) : 0.0f;
        o.AF[k].y = (m >= j1) ? onem * pow_int(a, a2, a4, a8, a16, m - j1) : 0.0f;
        o.AB[k].x = (m <= j0) ? onem * pow_int(a, a2, a4, a8, a16, j0 - m) : 0.0f;
        o.AB[k].y = (m <= j1) ? onem * pow_int(a, a2, a4, a8, a16, j1 - m) : 0.0f;
    }
#pragma unroll
    for (int r = 0; r < 8; ++r) {
        // D-matrix layout: VGPR r holds row m = r (lanes 0-15) / r+8 (lanes 16-31)
        const int mr = r + 8 * half;
        o.PF[r] = pow_int(a, a2, a4, a8, a16, mr + 1);
        o.QB[r] = pow_int(a, a2, a4, a8, a16, 16 - mr);
    }
    return o;
}

// Forward causal scan over a 256x16 LDS tile (pos-major [pos][16ch]).
// src may equal dst (each block fully read into D before being written back).
__device__ __forceinline__ void scan_fwd(const float* src, float* dst,
                                         const FiltOps& o, int lane) {
    const int half = lane >> 4, ch = lane & 15, ro = 2 * half;
    float carry = src[ch];                     // mirror border: carry0 = x[0][ch]
#pragma unroll 1
    for (int t = 0; t < 16; ++t) {
        const float* blk = src + t * 256;
        v8f d = {};
#pragma unroll
        for (int k = 0; k < 4; ++k) {
            // B 4x16 f32 layout: lanes 0-15 hold K=0,1 ; lanes 16-31 hold K=2,3
            v2f b;
            b.x = blk[(4 * k + ro) * 16 + ch];
            b.y = blk[(4 * k + ro + 1) * 16 + ch];
            d = __builtin_amdgcn_wmma_f32_16x16x4_f32(
                    false, o.AF[k], false, b, (short)0, d, false, false);
        }
        float* ob = dst + t * 256;
        float y7 = 0.0f;
#pragma unroll
        for (int r = 0; r < 8; ++r) {
            const float y = d[r] + o.PF[r] * carry;
            ob[(r + 8 * half) * 16 + ch] = y;
            if (r == 7) y7 = y;
        }
        // new carry = row 15 of this block = y7 of lanes 16..31, broadcast per channel
        carry = __int_as_float(
            __builtin_amdgcn_ds_bpermute((16 + ch) << 2, __float_as_int(y7)));
    }
}

// Backward anticausal scan, in place on buf. If ACC, accumulate scale*y into acc
// instead of writing back.
template <bool ACC>
__device__ __forceinline__ void scan_bwd(float* buf, float* acc, float scale,
                                         const FiltOps& o, int lane) {
    const int half = lane >> 4, ch = lane & 15, ro = 2 * half;
    float carry = buf[255 * 16 + ch];          // mirror border: carry = z[255][ch]
#pragma unroll 1
    for (int t = 15; t >= 0; --t) {
        float* blk = buf + t * 256;
        v8f d = {};
#pragma unroll
        for (int k = 0; k < 4; ++k) {
            v2f b;
            b.x = blk[(4 * k + ro) * 16 + ch];
            b.y = blk[(4 * k + ro + 1) * 16 + ch];
            d = __builtin_amdgcn_wmma_f32_16x16x4_f32(
                    false, o.AB[k], false, b, (short)0, d, false, false);
        }
        float y0 = 0.0f;
#pragma unroll
        for (int r = 0; r < 8; ++r) {
            const float y = d[r] + o.QB[r] * carry;
            const int idx = (r + 8 * half) * 16 + ch;
            if (ACC) acc[t * 256 + idx] += scale * y;
            else     blk[idx] = y;
            if (r == 0) y0 = y;
        }
        // new carry = row 0 of this block = y0 of lanes 0..15
        carry = __int_as_float(
            __builtin_amdgcn_ds_bpermute(ch << 2, __float_as_int(y0)));
    }
}

// ---------------------------------------------------------------------------
// Pass 1: scan along W for all 4 filters. One block per (n,h); 6 waves, one
// 16-channel chunk each. x row (256*96 f32, fully contiguous) is staged into
// LDS with async DMA. Output to ws in layout [f][n][cc][w][h][16] so pass 2
// loads are contiguous 16KB chunks (also via async DMA).
// ---------------------------------------------------------------------------
__global__ void __launch_bounds__(192) iir_w_pass(const float* __restrict__ x,
                                                  float* __restrict__ yh) {
    extern __shared__ float sm[];
    float* xs = sm;              // [NCC][TILE] input tiles (preserved across filters)
    float* ys = sm + NCC * TILE; // [NCC][TILE] scan workspace

    const int n   = blockIdx.x / HH;
    const int h   = blockIdx.x % HH;
    const int tid = threadIdx.x;

    // async stage of the full contiguous 98KB row: 6144 x b128
    const float* row = x + (size_t)(n * HH + h) * (WW * CCH);
    const uint32_t xs_base = lds_addr(xs);
#pragma unroll
    for (int i = 0; i < 32; ++i) {
        const int f4 = i * 192 + tid;
        const int w  = f4 / 24;              // 24 float4 per pixel (96 ch)
        const int c4 = f4 % 24;
        const uint32_t dst =
            xs_base + (uint32_t)(((c4 >> 2) * TILE + w * 16 + (c4 & 3) * 4) * 4);
        async_load_b128(dst, row, (uint32_t)(f4 * 16));
    }
    wait_async0();
    __syncthreads();

    const int lane = tid & 31;
    const int cc   = tid >> 5;               // wave id == channel chunk
    float* xt = xs + cc * TILE;
    float* yt = ys + cc * TILE;
    const uint32_t yt_base = lds_addr(yt);

    const float AV[4] = {0.1f, 0.3f, 0.4f, 0.8f};
#pragma unroll 1
    for (int f = 0; f < 4; ++f) {
        const FiltOps o = make_ops(AV[f], lane);
        scan_fwd(xt, yt, o, lane);
        scan_bwd<false>(yt, nullptr, 0.0f, o, lane);
        // async store tile to ws: [f][n][cc][w][h][16]; SGPR base is the
        // (f,n)-plane (uniform), wave-dependent cc folded into the vgpr offset.
        float* plane = yh + (((size_t)f * NN + n) * NCC) * WW * (size_t)(HH * 16);
#pragma unroll
        for (int i = 0; i < 32; ++i) {       // 1024 x b128 per tile
            const int f4 = i * 32 + lane;
            const int w  = f4 >> 2;
            const int c0 = (f4 & 3) * 4;
            const uint32_t voff =
                (uint32_t)((((cc * WW + w) * HH + h) * 16 + c0) * 4);
            async_store_b128(yt_base + (uint32_t)((w * 16 + c0) * 4), plane, voff);
        }
        wait_async0();   // yt is reused by the next filter
    }
}

// ---------------------------------------------------------------------------
// Pass 2: scan along H, weight by per-channel softmax, sum over filters.
// One block per (n,w); wave per channel chunk; per-filter 16KB chunk from ws
// is fully contiguous (async DMA in). Accumulate in LDS, then async NHWC store.
// ---------------------------------------------------------------------------
__global__ void __launch_bounds__(192) iir_h_pass(const float* __restrict__ yh,
                                                  const float* __restrict__ logits,
                                                  float* __restrict__ out) {
    extern __shared__ float sm[];
    float* work = sm;              // [NCC][TILE]
    float* acc  = sm + NCC * TILE; // [NCC][TILE]

    const int n   = blockIdx.x / WW;
    const int w   = blockIdx.x % WW;
    const int tid = threadIdx.x;
    const int lane = tid & 31;
    const int cc   = tid >> 5;
    float* wt = work + cc * TILE;
    float* at = acc + cc * TILE;
    const uint32_t wt_base = lds_addr(wt);

#pragma unroll
    for (int i = 0; i < 128; ++i) at[i * 32 + lane] = 0.0f;   // wave-private zero

    // per-channel softmax weights over the 4 filters
    const int cg = cc * 16 + (lane & 15);
    const float l0 = logits[0 * CCH + cg], l1 = logits[1 * CCH + cg];
    const float l2 = logits[2 * CCH + cg], l3 = logits[3 * CCH + cg];
    const float mx = fmaxf(fmaxf(l0, l1), fmaxf(l2, l3));
    const float e0 = __expf(l0 - mx), e1 = __expf(l1 - mx);
    const float e2 = __expf(l2 - mx), e3 = __expf(l3 - mx);
    const float inv = 1.0f / (e0 + e1 + e2 + e3);
    const float wgt[4] = {e0 * inv, e1 * inv, e2 * inv, e3 * inv};

    const float AV[4] = {0.1f, 0.3f, 0.4f, 0.8f};
#pragma unroll 1
    for (int f = 0; f < 4; ++f) {
        // async load of the contiguous 16KB (cc,w) chunk of plane (f,n)
        const float* plane =
            yh + (((size_t)f * NN + n) * NCC) * WW * (size_t)(HH * 16);
#pragma unroll
        for (int i = 0; i < 32; ++i) {
            const int f4 = i * 32 + lane;
            const uint32_t voff =
                (uint32_t)(((cc * WW + w) * (HH * 16) + f4 * 4) * 4);
            async_load_b128(wt_base + (uint32_t)(f4 * 16), plane, voff);
        }
        wait_async0();
        const FiltOps o = make_ops(AV[f], lane);
        scan_fwd(wt, wt, o, lane);                 // in-place vertical forward
        scan_bwd<true>(wt, at, wgt[f], o, lane);   // backward + weighted accumulate
    }
    __syncthreads();

    // async NHWC store: for each h, the 96 channels are 384B contiguous
    float* obase = out + ((size_t)(n * HH) * WW + w) * CCH;
    const uint32_t acc_base = lds_addr(acc);
#pragma unroll
    for (int i = 0; i < 32; ++i) {
        const int f4 = i * 192 + tid;        // 6144 x b128 = 256h * 96c
        const int h  = f4 / 24;
        const int c4 = f4 % 24;
        const uint32_t lsrc =
            acc_base + (uint32_t)((((c4 >> 2) * TILE) + h * 16 + (c4 & 3) * 4) * 4);
        const uint32_t voff = (uint32_t)((h * (WW * CCH) + c4 * 4) * 4);
        async_store_b128(lsrc, obase, voff);
    }
    // s_endpgm performs an implicit wait-idle, draining the async stores.
}

extern "C" void kernel_launch(void* const* d_in, const int* in_sizes, int n_in,
                              void* d_out, int out_size, void* d_ws, size_t ws_size,
                              hipStream_t stream) {
    (void)in_sizes; (void)n_in; (void)out_size; (void)ws_size;
    const float* x      = (const float*)d_in[0];
    const float* logits = (const float*)d_in[1];
    float* out = (float*)d_out;
    float* yh  = (float*)d_ws;   // needs 4 * 4*256*256*96 * 4B = 402.7 MB scratch

    const size_t shmem = (size_t)2 * NCC * TILE * sizeof(float);  // 192 KB
    iir_w_pass<<<NN * HH, 192, shmem, stream>>>(x, yh);
    iir_h_pass<<<NN * WW, 192, shmem, stream>>>(yh, logits, out);
}